// BSpline_50577534878013
// MI455X (gfx1250) — compile-verified
//
#include <hip/hip_runtime.h>
#include <stdint.h>

#define KNOTS 1024
#define DEG   3
#define NCTRL (KNOTS - DEG)              // 1021 control points
#define TPLEN (KNOTS + 2 * DEG)          // 1030 edge-padded knots
#define TABLEN (TPLEN + NCTRL)           // 2051 packed table entries
#define TABPAD ((TABLEN + 3) & ~3)       // 2052, multiple of 4 for b128 staging

typedef float __attribute__((ext_vector_type(4))) fv4;

// ---------- kernel 1: sort knots, edge-pad, pack [tp | c] into workspace ----------
__global__ void __launch_bounds__(KNOTS)
sort_pad_knots(const float* __restrict__ knots, const float* __restrict__ cg,
               float* __restrict__ tab) {
    __shared__ float s[KNOTS];
    const int t = threadIdx.x;
    s[t] = knots[t];
    __syncthreads();
    // bitonic sort, ascending
    for (int k = 2; k <= KNOTS; k <<= 1) {
        for (int j = k >> 1; j > 0; j >>= 1) {
            int ixj = t ^ j;
            if (ixj > t) {
                float a = s[t], b = s[ixj];
                bool asc = ((t & k) == 0);
                if ((a > b) == asc) { s[t] = b; s[ixj] = a; }
            }
            __syncthreads();
        }
    }
    // replicate-pad: tab[i] = sorted[clamp(i - DEG, 0, KNOTS-1)], i < TPLEN
    for (int i = t; i < TPLEN; i += KNOTS) {
        int src = i - DEG;
        src = src < 0 ? 0 : (src > KNOTS - 1 ? KNOTS - 1 : src);
        tab[i] = s[src];
    }
    // append control points
    for (int i = t; i < NCTRL; i += KNOTS)
        tab[TPLEN + i] = cg[i];
    if (t == 0)
        for (int i = TABLEN; i < TABPAD; ++i) tab[i] = 0.0f;  // staging pad
}

// rcp + one Newton step: ~0.5 ulp, 1 trans + 2 fma (vs slow IEEE divide expansion)
__device__ __forceinline__ float nr_rcp(float v) {
    float r = __builtin_amdgcn_rcpf(v);
    return fmaf(fmaf(-v, r, 1.0f), r, r);
}

__device__ __forceinline__ int wrap_nctrl(int i) {
    i = (i < 0) ? i + NCTRL : i;
    return (i >= NCTRL) ? i - NCTRL : i;
}

__device__ __forceinline__ float eval_one(float x,
                                          const float* __restrict__ s_tp,
                                          const float* __restrict__ s_c) {
    const float* ts = s_tp + DEG;               // sorted knots view
    // branchless lower_bound: cnt = #{ts < x}
    int cnt = 0;
#pragma unroll
    for (int step = KNOTS >> 1; step >= 1; step >>= 1)
        cnt += (ts[cnt + step - 1] < x) ? step : 0;

    const int k = cnt - 1 + DEG;                // torch: (x > t).sum() - 1 + p
    const int b = k - 2;                        // 6 contiguous padded knots tp[b..b+5]
    const float T0 = s_tp[b + 0], T1 = s_tp[b + 1], T2 = s_tp[b + 2];
    const float T3 = s_tp[b + 3], T4 = s_tp[b + 4], T5 = s_tp[b + 5];

    const int ci = k - 2 * DEG;                 // c[(k-2p+j) mod n]
    float d0 = s_c[wrap_nctrl(ci + 0)];
    float d1 = s_c[wrap_nctrl(ci + 1)];
    float d2 = s_c[wrap_nctrl(ci + 2)];
    float d3 = s_c[wrap_nctrl(ci + 3)];

    float a;
    // r = 1 (j = 3,2,1)
    a = (x - T2) * nr_rcp(T5 - T2); d3 = fmaf(a, d3 - d2, d2);
    a = (x - T1) * nr_rcp(T4 - T1); d2 = fmaf(a, d2 - d1, d1);
    a = (x - T0) * nr_rcp(T3 - T0); d1 = fmaf(a, d1 - d0, d0);
    // r = 2 (j = 3,2)
    a = (x - T2) * nr_rcp(T4 - T2); d3 = fmaf(a, d3 - d2, d2);
    a = (x - T1) * nr_rcp(T3 - T1); d2 = fmaf(a, d2 - d1, d1);
    // r = 3 (j = 3)
    a = (x - T2) * nr_rcp(T3 - T2); d3 = fmaf(a, d3 - d2, d2);
    return d3;
}

// -------------------- kernel 2: de Boor evaluation --------------------
__global__ void __launch_bounds__(256)
deboor_eval(const fv4* __restrict__ x4,
            const float* __restrict__ tab_g,
            fv4* __restrict__ out4,
            int n4, int rem, int ntot) {
    __shared__ __align__(16) float s_tab[TABPAD];
    const float* s_tp = s_tab;
    const float* s_c  = s_tab + TPLEN;
    const int tid = threadIdx.x;

    // Stage packed table into LDS via CDNA5 async DMA, 16B chunks (ASYNCcnt tracked).
    {
        const uint64_t base = (uint64_t)tab_g;
        const int nchunk = TABPAD / 4;          // 513
        for (int i = tid; i < nchunk; i += 256) {
            unsigned lds = (unsigned)(uintptr_t)&s_tab[4 * i];
            unsigned off = (unsigned)(i * 16);
            asm volatile("global_load_async_to_lds_b128 %0, %1, %2"
                         :: "v"(lds), "v"(off), "s"(base) : "memory");
        }
#if __has_builtin(__builtin_amdgcn_s_wait_asynccnt)
        __builtin_amdgcn_s_wait_asynccnt(0);
#else
        asm volatile("s_wait_asynccnt 0" ::: "memory");
#endif
    }
    __syncthreads();

    // grid-stride over float4 chunks; streamed once -> non-temporal
    const int stride = gridDim.x * 256;
    for (int i = blockIdx.x * 256 + tid; i < n4; i += stride) {
        fv4 v = __builtin_nontemporal_load(&x4[i]);
        fv4 o;
        o.x = eval_one(v.x, s_tp, s_c);
        o.y = eval_one(v.y, s_tp, s_c);
        o.z = eval_one(v.z, s_tp, s_c);
        o.w = eval_one(v.w, s_tp, s_c);
        __builtin_nontemporal_store(o, &out4[i]);
    }
    // scalar tail (ntot not divisible by 4)
    if (blockIdx.x == 0 && tid < rem) {
        const float* xs = (const float*)x4;
        float* os = (float*)out4;
        int idx = n4 * 4 + tid;
        if (idx < ntot) os[idx] = eval_one(xs[idx], s_tp, s_c);
    }
}

extern "C" void kernel_launch(void* const* d_in, const int* in_sizes, int n_in,
                              void* d_out, int out_size, void* d_ws, size_t ws_size,
                              hipStream_t stream) {
    (void)n_in; (void)ws_size; (void)out_size;
    const float* x     = (const float*)d_in[0];
    const float* knots = (const float*)d_in[1];
    const float* c     = (const float*)d_in[2];
    float* out = (float*)d_out;
    float* tab = (float*)d_ws;                   // TABPAD floats of scratch
    const int n = in_sizes[0];

    sort_pad_knots<<<1, KNOTS, 0, stream>>>(knots, c, tab);

    const int n4  = n >> 2;
    const int rem = n & 3;
    int blocks = (n4 + 4 * 256 - 1) / (4 * 256);   // 4 float4 per thread
    if (blocks < 1) blocks = 1;
    deboor_eval<<<blocks, 256, 0, stream>>>((const fv4*)x, tab,
                                            (fv4*)out, n4, rem, n);
}